// LocalNbrPool_53309134078324
// MI455X (gfx1250) — compile-verified
//
#include <hip/hip_runtime.h>
#include <hip/hip_bf16.h>
#include <stdint.h>

// Problem constants (match reference): B=8, N=512, F=128
#define BB 8
#define NN 512
#define FF 128
#define TI 16              // output rows (i) per block
#define TJ 64              // j tile staged in LDS per iteration
#define NTILES (NN / TJ)   // 8
#define NEGBIG (-1e10f)

typedef float v2f __attribute__((ext_vector_type(2)));

// out[b,i,f] = max_j ( X[b,j,f] + (A[b,j,i]!=0 ? 0 : -1e10) )
//
// Block: 128 threads = 4 wave32; lane -> feature f. Each wave keeps 16
// accumulators (one per local i) in VGPRs.
// X tiles: double-buffered LDS, filled with CDNA5 async global->LDS copies
// (global_load_async_to_lds_b128 / ASYNCcnt) overlapping compute.
// Inner loop (guaranteed by inline asm):
//   v_pk_add_f32    -> both candidates of a j-pair in one instruction
//   v_max3_num_f32  -> both candidates folded into acc in one instruction
// => 1 VALU instruction per element update (~8.4M total).
__global__ __launch_bounds__(128)
void LocalNbrPool_kernel(const float* __restrict__ X,
                         const float* __restrict__ A,
                         float* __restrict__ out) {
    __shared__ float Xs[2][TJ * FF];            // 2 x 32 KB
    // Pair-interleaved mask tile: Msp[buf][jp*32 + k*2 + (j&1)]
    __shared__ float Msp[2][(TJ / 2) * TI * 2]; // 2 x 4 KB

    const int tid = threadIdx.x;
    const int b   = blockIdx.x / (NN / TI);
    const int i0  = (blockIdx.x % (NN / TI)) * TI;
    const int f   = tid & 127;

    // Mask-staging role for this thread: half a row of A per tile.
    const int jl   = tid >> 1;   // local j row 0..63
    const int half = tid & 1;    // which 8 of the 16 i's
    const int jp_s = jl >> 1;    // j-pair index
    const int sub  = jl & 1;     // position within the pair

    float acc[TI];
#pragma unroll
    for (int k = 0; k < TI; ++k) acc[k] = -__builtin_inff();

    const unsigned ldsXs0 = (unsigned)(uint64_t)(uintptr_t)(void*)&Xs[0][0];

    // ---- helper lambdas ----
    auto issue_async_x = [&](int tile, int buf) {
        const float* xbase = X + (size_t)(b * NN + tile * TJ) * FF;
        const unsigned ldsb = ldsXs0 + (unsigned)buf * (TJ * FF * 4);
#pragma unroll
        for (int r = 0; r < 16; ++r) {
            unsigned off  = (unsigned)(tid * 16 + r * (128 * 16)); // byte offset
            unsigned ldsa = ldsb + off;
            asm volatile("global_load_async_to_lds_b128 %0, %1, %2"
                         :
                         : "v"(ldsa), "v"(off), "s"(xbase)
                         : "memory");
        }
    };
    auto load_a = [&](int tile, float4& a0, float4& a1) {
        const float4* arow =
            (const float4*)(A + (size_t)(b * NN + tile * TJ + jl) * NN + i0) + half * 2;
        a0 = arow[0];
        a1 = arow[1];
    };
    auto store_m = [&](int buf, const float4& a0, const float4& a1) {
        float m[8];
        m[0] = (a0.x == 0.0f) ? NEGBIG : 0.0f;
        m[1] = (a0.y == 0.0f) ? NEGBIG : 0.0f;
        m[2] = (a0.z == 0.0f) ? NEGBIG : 0.0f;
        m[3] = (a0.w == 0.0f) ? NEGBIG : 0.0f;
        m[4] = (a1.x == 0.0f) ? NEGBIG : 0.0f;
        m[5] = (a1.y == 0.0f) ? NEGBIG : 0.0f;
        m[6] = (a1.z == 0.0f) ? NEGBIG : 0.0f;
        m[7] = (a1.w == 0.0f) ? NEGBIG : 0.0f;
#pragma unroll
        for (int q = 0; q < 8; ++q) {
            const int k = half * 8 + q;
            Msp[buf][jp_s * (TI * 2) + k * 2 + sub] = m[q];
        }
    };

    // ---- prologue: stage tile 0 into buffer 0 ----
    issue_async_x(0, 0);
    {
        float4 a0, a1;
        load_a(0, a0, a1);
        store_m(0, a0, a1);
    }
    asm volatile("s_wait_asynccnt 0" ::: "memory");
    __syncthreads();

    // ---- main pipeline ----
    for (int t = 0; t < NTILES; ++t) {
        const int cur  = t & 1;
        const int nxt  = cur ^ 1;
        const bool pre = (t + 1) < NTILES;

        float4 a0, a1;
        if (pre) {
            issue_async_x(t + 1, nxt);  // async copy overlaps compute below
            load_a(t + 1, a0, a1);      // results consumed after compute
        }

        // ---- compute tile t: v_pk_add_f32 + v_max3_num_f32 ----
        const float* xsb = &Xs[cur][0];
        const v2f*   mpb = (const v2f*)&Msp[cur][0];
#pragma unroll 2
        for (int jp = 0; jp < TJ / 2; ++jp) {
            v2f x2;
            x2.x = xsb[(2 * jp + 0) * FF + f];   // ds_load_b32 (merge to 2addr)
            x2.y = xsb[(2 * jp + 1) * FF + f];
            const v2f* mrow = mpb + jp * TI;     // 16 packed pairs (broadcast)
#pragma unroll
            for (int k = 0; k < TI; ++k) {
                const v2f m = mrow[k];
                v2f c;
                asm("v_pk_add_f32 %0, %1, %2"
                    : "=v"(c)
                    : "v"(x2), "v"(m));
                float r;
                asm("v_max3_num_f32 %0, %1, %2, %3"
                    : "=v"(r)
                    : "v"(acc[k]), "v"(c.x), "v"(c.y));
                acc[k] = r;
            }
        }

        if (pre) {
            store_m(nxt, a0, a1);
            asm volatile("s_wait_asynccnt 0" ::: "memory");
        }
        __syncthreads();
    }

    // ---- write out[b, i0+k, f] ----
    float* orow = out + (size_t)(b * NN + i0) * FF + f;
#pragma unroll
    for (int k = 0; k < TI; ++k)
        orow[(size_t)k * FF] = acc[k];
}

extern "C" void kernel_launch(void* const* d_in, const int* in_sizes, int n_in,
                              void* d_out, int out_size, void* d_ws, size_t ws_size,
                              hipStream_t stream) {
    const float* X = (const float*)d_in[0];   // [B, N, F] float32
    const float* A = (const float*)d_in[1];   // [B, N, N] float32
    float* out     = (float*)d_out;           // [B, N, F] float32
    (void)in_sizes; (void)n_in; (void)out_size; (void)d_ws; (void)ws_size;

    dim3 grid(BB * (NN / TI));   // 256 blocks
    dim3 block(128);             // 4 wave32
    hipLaunchKernelGGL(LocalNbrPool_kernel, grid, block, 0, stream, X, A, out);
}